// RelativePositionEncoding_86371792322629
// MI455X (gfx1250) — compile-verified
//
#include <hip/hip_runtime.h>

namespace {
constexpr int kRMax      = 32;
constexpr int kSMax      = 2;
constexpr int kResBins   = 2 * kRMax + 2;                  // 66
constexpr int kChainBins = 2 * kSMax + 2;                  // 6
constexpr int kNoBins    = 2 * kResBins + 1 + kChainBins;  // 139
constexpr int kCz        = 128;
constexpr int kJChunk    = 1024;
constexpr int kThreads   = 256;
}

typedef float v4f __attribute__((ext_vector_type(4)));

__global__ __launch_bounds__(kThreads)
void relpos_enc_kernel(const int* __restrict__ asym_id,
                       const int* __restrict__ residue_index,
                       const int* __restrict__ entity_id,
                       const int* __restrict__ token_index,
                       const int* __restrict__ sym_id,
                       const float* __restrict__ W,   // [kCz][kNoBins] row-major
                       float* __restrict__ out,       // [N][N][kCz]
                       int N)
{
  __shared__ float    sW[kNoBins * kCz];   // transposed: sW[bin * kCz + c]
  __shared__ unsigned meta[kJChunk];

  const int tid  = threadIdx.x;
  const int i    = blockIdx.x;
  const int j0   = blockIdx.y * kJChunk;
  const int jcnt = (N - j0 < kJChunk) ? (N - j0) : kJChunk;

  // ---- Phase 1: async, transposing DMA of W into LDS (ASYNCcnt path) ----
  // Flat shared addresses: low 32 bits == LDS byte offset (verified: compiler
  // folded base 0 for sW in round-1 asm).
  {
    const unsigned ldsBase = (unsigned)(size_t)(&sW[0]);
    for (int idx = tid; idx < kCz * kNoBins; idx += kThreads) {
      const int c = idx / kNoBins;
      const int b = idx - c * kNoBins;
      const unsigned ldsAddr = ldsBase + (unsigned)((b * kCz + c) * 4);
      const unsigned long long gaddr = (unsigned long long)(size_t)(W + idx);
      asm volatile("global_load_async_to_lds_b32 %0, %1, off"
                   :: "v"(ldsAddr), "v"(gaddr)
                   : "memory");
    }
  }

  // ---- Phase 2: per-j packed bin metadata for this row i (overlaps DMA) ----
  const int ai = asym_id[i];
  const int ri = residue_index[i];
  const int ei = entity_id[i];
  const int ti = token_index[i];
  const int si = sym_id[i];

  for (int jj = tid; jj < jcnt; jj += kThreads) {
    const int j  = j0 + jj;
    const int aj = asym_id[j];
    const int rj = residue_index[j];
    const int ej = entity_id[j];
    const int tj = token_index[j];
    const int sj = sym_id[j];

    const bool same_chain  = (ai == aj);
    const bool same_res    = (ri == rj);
    const bool same_entity = (ei == ej);

    int dres = ri - rj + kRMax;
    dres = dres < 0 ? 0 : (dres > 2 * kRMax ? 2 * kRMax : dres);
    const int binA = same_chain ? dres : (2 * kRMax + 1);

    int dtok = ti - tj + kRMax;
    dtok = dtok < 0 ? 0 : (dtok > 2 * kRMax ? 2 * kRMax : dtok);
    const int binB = kResBins + ((same_chain && same_res) ? dtok : (2 * kRMax + 1));

    int dch = si - sj + kSMax;
    dch = dch < 0 ? 0 : (dch > 2 * kSMax ? 2 * kSMax : dch);
    const int binC = 2 * kResBins + 1 + (same_entity ? dch : (2 * kSMax + 1));

    meta[jj] = (unsigned)binA | ((unsigned)binB << 8) | ((unsigned)binC << 16) |
               (same_entity ? (1u << 24) : 0u);
  }

  asm volatile("s_wait_asynccnt 0" ::: "memory");
  __syncthreads();

  // ---- Phase 3: one wave per pair; lane owns 4 consecutive channels ----
  const int lane = tid & 31;          // wave32
  const int wave = tid >> 5;
  constexpr int kWaves = kThreads / 32;

  const v4f* __restrict__ sW4 = (const v4f*)sW;
  const v4f sSe = sW4[(2 * kResBins) * (kCz / 4) + lane];   // column 132, loop-invariant

  float* const outRow = out + ((size_t)i * (size_t)N + (size_t)j0) * kCz + lane * 4;

#pragma unroll 4
  for (int jj = wave; jj < jcnt; jj += kWaves) {
    const unsigned m = meta[jj];                 // same-address LDS broadcast
    const int  binA =  m        & 0xFF;
    const int  binB = (m >> 8)  & 0xFF;
    const int  binC = (m >> 16) & 0xFF;
    const float se  = (m >> 24) ? 1.0f : 0.0f;

    const v4f a = sW4[binA * (kCz / 4) + lane];  // ds_load_b128, conflict-free
    const v4f b = sW4[binB * (kCz / 4) + lane];
    const v4f c = sW4[binC * (kCz / 4) + lane];

    v4f o;
    o.x = fmaf(se, sSe.x, a.x + b.x + c.x);
    o.y = fmaf(se, sSe.y, a.y + b.y + c.y);
    o.z = fmaf(se, sSe.z, a.z + b.z + c.z);
    o.w = fmaf(se, sSe.w, a.w + b.w + c.w);

    // Streaming output (512 MiB, write-once): non-temporal 512B/wave store,
    // keeps L2 free for the shared W and index arrays.
    __builtin_nontemporal_store(o, (v4f*)(outRow + (size_t)jj * kCz));
  }
}

extern "C" void kernel_launch(void* const* d_in, const int* in_sizes, int n_in,
                              void* d_out, int out_size, void* d_ws, size_t ws_size,
                              hipStream_t stream) {
  const int*   asym = (const int*)d_in[0];
  const int*   resi = (const int*)d_in[1];
  const int*   enti = (const int*)d_in[2];
  const int*   toki = (const int*)d_in[3];
  const int*   symi = (const int*)d_in[4];
  const float* W    = (const float*)d_in[5];
  float*       out  = (float*)d_out;

  const int N = in_sizes[0];   // B == 1 in the reference setup
  dim3 grid(N, (N + kJChunk - 1) / kJChunk);
  relpos_enc_kernel<<<grid, kThreads, 0, stream>>>(asym, resi, enti, toki, symi, W, out, N);
}